// NeuralODE_15822659518760
// MI455X (gfx1250) — compile-verified
//
#include <hip/hip_runtime.h>
#include <cmath>

// ---------------------------------------------------------------------------
// Syrinx Van-der-Pol-like ODE, Tsit5 fixed steps on the save grid.
//   Kernel 1 (parallel, fills GPU): precompute eps/B/D0 at all 6 stage times
//             of every step -> 20-float record per step in d_ws (1.92 MB).
//   Kernel 2 (serial scan): 512 threads = 16 wave32 waves, one wave per block
//             so each wave owns a SIMD; pure FMA chain + one b128-burst
//             uniform table read per step + global_prefetch_b8 lookahead.
// ---------------------------------------------------------------------------

namespace {

// Tsit5 tableau (node fractions C, stage weights A, output weights W)
constexpr float TC2 = 0.161f, TC3 = 0.327f, TC4 = 0.9f, TC5 = 0.9800255409045097f;
constexpr float A21 = 0.161f;
constexpr float A31 = -0.008480655492356989f, A32 = 0.335480655492357f;
constexpr float A41 = 2.8971530571054935f, A42 = -6.359448489975075f, A43 = 4.3622954328695815f;
constexpr float A51 = 5.325864828439257f, A52 = -11.748883564062828f,
                A53 = 7.4955393428898365f, A54 = -0.09249506636175525f;
constexpr float A61 = 5.86145544294642f, A62 = -12.92096931784711f,
                A63 = 8.159367898576159f, A64 = -0.071584973281401f,
                A65 = -0.028269050394068383f;
constexpr float W1 = 0.09646076681806523f, W2 = 0.01f, W3 = 0.4798896504144996f,
                W4 = 1.379008574103742f, W5 = -3.290069515436081f, W6 = 2.324710524099774f;

// vdot = -eps*x - Cc*x^2*v + B*v - D0  ==  (B - Cc*x^2)*v - eps*x - D0
__device__ __forceinline__ float fv(float eps, float Bv, float D0, float Cc,
                                    float x, float v) {
    return fmaf(fmaf(-Cc, x * x, Bv), v, fmaf(-eps, x, -D0));
}

} // namespace

// --- Kernel 1: per-step coefficient table -----------------------------------
// Record layout (20 floats, 80 B, 16B-aligned): [h, (eps,B,D0) x 6 stages, pad]
__global__ void syrinx_coef_kernel(const float* __restrict__ ts,
                                   const float* __restrict__ params,
                                   float* __restrict__ tab,
                                   int nstep, float knorm) {
    int i = blockIdx.x * blockDim.x + threadIdx.x;
    if (i >= nstep) return;

    float t0 = ts[i];
    float h  = ts[i + 1] - t0;
    float p0 = params[0], p1 = params[1], p2 = params[2], p3 = params[3], p5 = params[5];

    const float stc[6] = {0.0f, TC2, TC3, TC4, TC5, 1.0f};
    float* o = tab + (size_t)i * 20;
    o[0]  = h;
    o[19] = 0.0f;

#pragma unroll
    for (int s = 0; s < 6; ++s) {
        float t = fmaf(stc[s], h, t0);

        // K_sum: KNORM * exp((t-l)/0.025) * max(l-t,0)^5, l in {0,0.5,1.0,1.5}
        const float kl[4] = {0.0f, 0.5f, 1.0f, 1.5f};
        float Ks = 0.0f;
#pragma unroll
        for (int j = 0; j < 4; ++j) {
            float m  = fmaxf(kl[j] - t, 0.0f);
            float m2 = m * m;
            float m5 = m2 * m2 * m;
            Ks += knorm * expf((t - kl[j]) / 0.025f) * m5;
        }

        // D_sum: DA[j]*exp(-0.5*(t-dl)^2/1e-4)
        const float dl[5] = {0.1f, 0.45f, 0.6f, 0.95f, 1.1f};
        const float da[5] = {0.05f, 0.02f, 0.01f, 0.05f, 0.03f};
        float Ds = 0.0f;
#pragma unroll
        for (int j = 0; j < 5; ++j) {
            float d = t - dl[j];
            Ds += da[j] * expf(-0.5f * d * d / 1.0e-4f);
        }

        // P_sum: sum over {0.3, 0.9} of (0.025*exp(-0.5*(t-p)^2/0.0064) - 0.005)
        const float pl[2] = {0.3f, 0.9f};
        float Ps = 0.0f;
#pragma unroll
        for (int j = 0; j < 2; ++j) {
            float d = t - pl[j];
            Ps += fmaf(0.025f, expf(-0.5f * d * d / 6.4e-3f), -0.005f);
        }

        float eps = (p0 + p1 * Ks) * 1.0e8f;
        float Bv  = (p2 + p3 * Ps) * 1.0e3f;
        float D0  = p5 * Ds * 1.0e7f;
        o[1 + 3 * s] = eps;
        o[2 + 3 * s] = Bv;
        o[3 + 3 * s] = D0;
    }
}

// --- Kernel 2: serial Tsit5 scan, one thread per trajectory -----------------
__global__ void __launch_bounds__(32, 1)
syrinx_integrate_kernel(const float* __restrict__ y0,
                        const float* __restrict__ params,
                        const float* __restrict__ tab,
                        float* __restrict__ out,
                        int nT, int batch) {
    int b = blockIdx.x * blockDim.x + threadIdx.x;
    if (b >= batch) return;

    const float Cc = params[4] * 1.0e8f;
    float x = y0[2 * b + 0];
    float v = y0[2 * b + 1];

    float2* ob = reinterpret_cast<float2*>(out + (size_t)b * (size_t)nT * 2);
    ob[0] = make_float2(x, v);

    const float4* __restrict__ q = reinterpret_cast<const float4*>(tab);
    const int nstep = nT - 1;

    for (int i = 0; i < nstep; ++i) {
        const float4* c = q + (size_t)i * 5;
        // gfx1250 prefetch: pull the table ~24 iterations (1.9 KB) ahead into WGP$
        __builtin_prefetch((const void*)(c + 5 * 24), 0, 1);

        float4 a0 = c[0], a1 = c[1], a2 = c[2], a3 = c[3], a4 = c[4];
        float h  = a0.x;
        float e1 = a0.y, b1 = a0.z, d1 = a0.w;
        float e2 = a1.x, b2 = a1.y, d2 = a1.z;
        float e3 = a1.w, b3 = a2.x, d3 = a2.y;
        float e4 = a2.z, b4 = a2.w, d4 = a3.x;
        float e5 = a3.y, b5 = a3.z, d5 = a3.w;
        float e6 = a4.x, b6 = a4.y, d6 = a4.z;

        float k1x = v;
        float k1v = fv(e1, b1, d1, Cc, x, v);

        float x2 = fmaf(h, A21 * k1x, x);
        float v2 = fmaf(h, A21 * k1v, v);
        float k2x = v2;
        float k2v = fv(e2, b2, d2, Cc, x2, v2);

        float s3x = fmaf(A32, k2x, A31 * k1x);
        float s3v = fmaf(A32, k2v, A31 * k1v);
        float x3 = fmaf(h, s3x, x);
        float v3 = fmaf(h, s3v, v);
        float k3x = v3;
        float k3v = fv(e3, b3, d3, Cc, x3, v3);

        float s4x = fmaf(A43, k3x, fmaf(A42, k2x, A41 * k1x));
        float s4v = fmaf(A43, k3v, fmaf(A42, k2v, A41 * k1v));
        float x4 = fmaf(h, s4x, x);
        float v4 = fmaf(h, s4v, v);
        float k4x = v4;
        float k4v = fv(e4, b4, d4, Cc, x4, v4);

        float s5x = fmaf(A54, k4x, fmaf(A53, k3x, fmaf(A52, k2x, A51 * k1x)));
        float s5v = fmaf(A54, k4v, fmaf(A53, k3v, fmaf(A52, k2v, A51 * k1v)));
        float x5 = fmaf(h, s5x, x);
        float v5 = fmaf(h, s5v, v);
        float k5x = v5;
        float k5v = fv(e5, b5, d5, Cc, x5, v5);

        float s6x = fmaf(A65, k5x, fmaf(A64, k4x, fmaf(A63, k3x, fmaf(A62, k2x, A61 * k1x))));
        float s6v = fmaf(A65, k5v, fmaf(A64, k4v, fmaf(A63, k3v, fmaf(A62, k2v, A61 * k1v))));
        float x6 = fmaf(h, s6x, x);
        float v6 = fmaf(h, s6v, v);
        float k6x = v6;
        float k6v = fv(e6, b6, d6, Cc, x6, v6);

        float sbx = fmaf(W6, k6x, fmaf(W5, k5x, fmaf(W4, k4x, fmaf(W3, k3x, fmaf(W2, k2x, W1 * k1x)))));
        float sbv = fmaf(W6, k6v, fmaf(W5, k5v, fmaf(W4, k4v, fmaf(W3, k3v, fmaf(W2, k2v, W1 * k1v)))));
        x = fmaf(h, sbx, x);
        v = fmaf(h, sbv, v);

        ob[i + 1] = make_float2(x, v);
    }
}

extern "C" void kernel_launch(void* const* d_in, const int* in_sizes, int n_in,
                              void* d_out, int out_size, void* d_ws, size_t ws_size,
                              hipStream_t stream) {
    (void)n_in; (void)out_size; (void)ws_size;
    const float* ts     = (const float*)d_in[0];   // (N_T,)
    const float* y0     = (const float*)d_in[1];   // (BATCH, 2)
    const float* params = (const float*)d_in[2];   // (6,)
    float* out = (float*)d_out;                    // (BATCH, N_T, 2)

    const int nT    = in_sizes[0];
    const int batch = in_sizes[1] / 2;
    const int nstep = nT - 1;

    float* tab = (float*)d_ws;   // nstep * 80 bytes (~1.92 MB) — lives in L2

    // KNORM computed in float64 exactly like the numpy reference, then cast.
    const double kn = 0.06 * std::exp(5.0 - 5.0 * std::log(5.0) - 5.0 * std::log(0.025));

    syrinx_coef_kernel<<<dim3((nstep + 255) / 256), dim3(256), 0, stream>>>(
        ts, params, tab, nstep, (float)kn);

    // One wave32 per block: 16 blocks spread across WGPs -> one wave per SIMD.
    syrinx_integrate_kernel<<<dim3((batch + 31) / 32), dim3(32), 0, stream>>>(
        y0, params, tab, out, nT, batch);
}